// MultiHeadAttention_49340584296922
// MI455X (gfx1250) — compile-verified
//
#include <hip/hip_runtime.h>
#include <hip/hip_bf16.h>

typedef __attribute__((ext_vector_type(2))) float v2f;
typedef __attribute__((ext_vector_type(8))) float v8f;

#define B_  4
#define S_  1024
#define D_  1024   // IN_FEATURES == H_*DO_
#define H_  16
#define DO_ 64

__device__ __forceinline__ v8f wmma_f32(v2f a, v2f b, v8f c) {
  // V_WMMA_F32_16X16X4_F32 : D = A(16x4) * B(4x16) + C(16x16), fp32 throughout
  return __builtin_amdgcn_wmma_f32_16x16x4_f32(
      /*neg_a=*/false, a, /*neg_b=*/false, b,
      /*c_mod=*/(short)0, c, /*reuse_a=*/false, /*reuse_b=*/false);
}

// ---------------------------------------------------------------------------
// C[m,n] = X[m,:] @ W[:,n] + bias[n];  X:[4096,1024], W:[1024,1024]
// Each wave: one 32x64 output tile (two A fragments share 4 B fragments:
// 8 WMMA / k-step, 10.7 flop/byte from L2).
// head_split=1 stores to [B,H,S,64]; else row-major [4096,1024].
// ---------------------------------------------------------------------------
__global__ __launch_bounds__(128) void gemm_xw(
    const float* __restrict__ X, const float* __restrict__ W,
    const float* __restrict__ bias, float* __restrict__ out, int head_split) {
  const int lane = threadIdx.x & 31;
  const int wave = threadIdx.x >> 5;
  const int m0   = blockIdx.x * 32;
  const int n0   = blockIdx.y * 256 + wave * 64;
  const int mr   = lane & 15;
  const int koff = (lane >> 4) * 2;        // K pair: lanes<16 -> {0,1}, lanes>=16 -> {2,3}
  const float* arow0 = X + (size_t)(m0 + mr) * D_;
  const float* arow1 = arow0 + 16 * D_;

  v8f acc[8] = {};
  for (int k0 = 0; k0 < D_; k0 += 4) {
    v2f a0 = *(const v2f*)(arow0 + k0 + koff);
    v2f a1 = *(const v2f*)(arow1 + k0 + koff);
    const float* wrow = W + (size_t)(k0 + koff) * D_;
#pragma unroll
    for (int t = 0; t < 4; ++t) {
      const int col = n0 + t * 16 + mr;
      v2f b;
      b.x = wrow[col];
      b.y = wrow[D_ + col];
      acc[t]     = wmma_f32(a0, b, acc[t]);
      acc[4 + t] = wmma_f32(a1, b, acc[4 + t]);
    }
  }
#pragma unroll
  for (int half = 0; half < 2; ++half) {
#pragma unroll
    for (int t = 0; t < 4; ++t) {
      const int col = n0 + t * 16 + mr;
      const float bb = bias[col];
#pragma unroll
      for (int r = 0; r < 8; ++r) {
        const int row = m0 + half * 16 + r + (lane >> 4) * 8;
        const float val = acc[half * 4 + t][r] + bb;
        if (head_split) {
          const int bi = row >> 10, s = row & 1023, h = col >> 6, d = col & 63;
          out[(((size_t)(bi * H_ + h) * S_) + s) * DO_ + d] = val;
        } else {
          out[(size_t)row * D_ + col] = val;
        }
      }
    }
  }
}

// ---------------------------------------------------------------------------
// Raw logits: attn[bh,q,k] = Qh[bh,q,:] . Kh[bh,k,:]   (K-dim = 64)
// 32x64 tile per wave; B fragments (Kh^T) shared across both q-halves.
// ---------------------------------------------------------------------------
__global__ __launch_bounds__(128) void attn_scores(
    const float* __restrict__ Qh, const float* __restrict__ Kh,
    float* __restrict__ attn) {
  const int lane = threadIdx.x & 31;
  const int wave = threadIdx.x >> 5;
  const int bh   = blockIdx.z;
  const int q0   = blockIdx.x * 32;
  const int n0   = blockIdx.y * 256 + wave * 64;
  const int mr   = lane & 15;
  const int koff = (lane >> 4) * 2;
  const float* qrow0 = Qh + ((size_t)bh * S_ + (q0 + mr)) * DO_;
  const float* qrow1 = qrow0 + 16 * DO_;
  const float* kbase = Kh + (size_t)bh * S_ * DO_;

  v8f acc[8] = {};
#pragma unroll
  for (int kk = 0; kk < DO_; kk += 4) {
    v2f a0 = *(const v2f*)(qrow0 + kk + koff);
    v2f a1 = *(const v2f*)(qrow1 + kk + koff);
#pragma unroll
    for (int t = 0; t < 4; ++t) {
      const int col = n0 + t * 16 + mr;
      v2f b = *(const v2f*)(kbase + (size_t)col * DO_ + kk + koff);  // (Kh^T) fragment
      acc[t]     = wmma_f32(a0, b, acc[t]);
      acc[4 + t] = wmma_f32(a1, b, acc[4 + t]);
    }
  }
  float* abase = attn + (size_t)bh * S_ * S_;
#pragma unroll
  for (int half = 0; half < 2; ++half) {
#pragma unroll
    for (int t = 0; t < 4; ++t) {
#pragma unroll
      for (int r = 0; r < 8; ++r) {
        const int qq = q0 + half * 16 + r + (lane >> 4) * 8;
        abase[(size_t)qq * S_ + n0 + t * 16 + mr] = acc[half * 4 + t][r];  // raw
      }
    }
  }
}

// ---------------------------------------------------------------------------
// Per-row masked softmax stats: m = max(valid ? 0.125*lg : -1e9), den = sum exp(s-m)
// One wave (32 lanes x 32 elems in registers) per (b,h,q) row.
// ---------------------------------------------------------------------------
__global__ __launch_bounds__(256) void softmax_stats(
    const float* __restrict__ attn, const float* __restrict__ q_mask,
    const float* __restrict__ k_mask, float* __restrict__ rowmax,
    float* __restrict__ rowsum) {
  const int lane = threadIdx.x & 31;
  const int wave = threadIdx.x >> 5;
  const int gid  = blockIdx.x * 8 + wave;     // (b*H+h)*S + q
  const int bh   = gid >> 10;
  const int q    = gid & 1023;
  const int b    = bh >> 4;
  const float qm = q_mask[b * S_ + q];
  const float* row = attn + (size_t)gid * S_;
  const float* km  = k_mask + b * S_;

  float vals[32];
  float m = -1e30f;
#pragma unroll
  for (int i = 0; i < 32; ++i) {
    const int kk = lane + i * 32;
    const bool valid = (qm != 0.0f) && (km[kk] != 0.0f);
    const float s = valid ? row[kk] * 0.125f : -1e9f;
    vals[i] = s;
    m = fmaxf(m, s);
  }
  for (int off = 16; off; off >>= 1) m = fmaxf(m, __shfl_xor(m, off, 32));
  float den = 0.0f;
#pragma unroll
  for (int i = 0; i < 32; ++i) den += __expf(vals[i] - m);   // exp(-1e9-m) -> 0
  for (int off = 16; off; off >>= 1) den += __shfl_xor(den, off, 32);
  if (lane == 0) { rowmax[gid] = m; rowsum[gid] = den; }
}

// ---------------------------------------------------------------------------
// ctx[b,q,h*64+d] = sum_k probs[q,k] * Vh[bh,k,d]
// probs recomputed per-lane directly in the WMMA A-fragment layout (no LDS).
// 32 q-rows per wave; V fragments shared across both q-halves.
// ---------------------------------------------------------------------------
__global__ __launch_bounds__(128) void attn_context(
    const float* __restrict__ attn, const float* __restrict__ Vh,
    const float* __restrict__ q_mask, const float* __restrict__ k_mask,
    const float* __restrict__ rowmax, const float* __restrict__ rowsum,
    float* __restrict__ ctx) {
  const int lane = threadIdx.x & 31;
  const int wave = threadIdx.x >> 5;
  const int bh   = blockIdx.z;
  const int b    = bh >> 4;
  const int h    = bh & 15;
  const int q0   = (blockIdx.x * 4 + wave) * 32;
  const int mr   = lane & 15;
  const int koff = (lane >> 4) * 2;
  const int qr0  = q0 + mr;
  const int qr1  = qr0 + 16;
  const int rid0 = bh * S_ + qr0;
  const int rid1 = bh * S_ + qr1;
  const float m0_  = rowmax[rid0];
  const float m1_  = rowmax[rid1];
  const float d0   = rowsum[rid0];
  const float d1   = rowsum[rid1];
  const float inv0 = (d0 > 0.0f) ? 1.0f / d0 : 0.0f;
  const float inv1 = (d1 > 0.0f) ? 1.0f / d1 : 0.0f;
  const float qm0  = q_mask[b * S_ + qr0];
  const float qm1  = q_mask[b * S_ + qr1];
  const float* km    = k_mask + b * S_;
  const float* lrow0 = attn + (size_t)rid0 * S_;
  const float* lrow1 = attn + (size_t)rid1 * S_;
  const float* vbase = Vh + (size_t)bh * S_ * DO_;

  v8f acc[8] = {};
  for (int k0 = 0; k0 < S_; k0 += 4) {
    const int kk = k0 + koff;
    const float kmx = km[kk];
    const float kmy = km[kk + 1];
    v2f lg0 = *(const v2f*)(lrow0 + kk);
    v2f lg1 = *(const v2f*)(lrow1 + kk);
    v2f a0, a1;
    a0.x = (qm0 != 0.0f && kmx != 0.0f) ? __expf(lg0.x * 0.125f - m0_) * inv0 : 0.0f;
    a0.y = (qm0 != 0.0f && kmy != 0.0f) ? __expf(lg0.y * 0.125f - m0_) * inv0 : 0.0f;
    a1.x = (qm1 != 0.0f && kmx != 0.0f) ? __expf(lg1.x * 0.125f - m1_) * inv1 : 0.0f;
    a1.y = (qm1 != 0.0f && kmy != 0.0f) ? __expf(lg1.y * 0.125f - m1_) * inv1 : 0.0f;
    const float* vrow = vbase + (size_t)kk * DO_;
#pragma unroll
    for (int t = 0; t < 4; ++t) {
      const int col = t * 16 + mr;
      v2f bf;
      bf.x = vrow[col];
      bf.y = vrow[DO_ + col];
      acc[t]     = wmma_f32(a0, bf, acc[t]);
      acc[4 + t] = wmma_f32(a1, bf, acc[4 + t]);
    }
  }
#pragma unroll
  for (int half = 0; half < 2; ++half) {
#pragma unroll
    for (int t = 0; t < 4; ++t) {
#pragma unroll
      for (int r = 0; r < 8; ++r) {
        const int qq  = q0 + half * 16 + r + (lane >> 4) * 8;
        const int col = t * 16 + mr;
        ctx[((size_t)(b * S_ + qq) * D_) + h * DO_ + col] = acc[half * 4 + t][r];
      }
    }
  }
}

// ---------------------------------------------------------------------------
// out = LayerNorm(out + residual) * gamma + beta, one wave per row.
// ---------------------------------------------------------------------------
__global__ __launch_bounds__(256) void residual_ln(
    float* __restrict__ out, const float* __restrict__ resid,
    const float* __restrict__ gamma, const float* __restrict__ beta) {
  const int lane = threadIdx.x & 31;
  const int wave = threadIdx.x >> 5;
  const int row  = blockIdx.x * 8 + wave;
  float* orow = out + (size_t)row * D_;
  const float* rrow = resid + (size_t)row * D_;

  float vals[32];
  float s = 0.0f, s2 = 0.0f;
#pragma unroll
  for (int i = 0; i < 32; ++i) {
    const int idx = lane + i * 32;
    const float x = orow[idx] + rrow[idx];
    vals[i] = x;
    s += x;
    s2 += x * x;
  }
  for (int off = 16; off; off >>= 1) {
    s  += __shfl_xor(s, off, 32);
    s2 += __shfl_xor(s2, off, 32);
  }
  const float mean = s * (1.0f / D_);
  const float var  = s2 * (1.0f / D_) - mean * mean;
  const float rstd = rsqrtf(var + 1e-5f);
#pragma unroll
  for (int i = 0; i < 32; ++i) {
    const int idx = lane + i * 32;
    orow[idx] = (vals[i] - mean) * rstd * gamma[idx] + beta[idx];
  }
}

// ---------------------------------------------------------------------------
extern "C" void kernel_launch(void* const* d_in, const int* in_sizes, int n_in,
                              void* d_out, int out_size, void* d_ws, size_t ws_size,
                              hipStream_t stream) {
  const float* q      = (const float*)d_in[0];
  const float* k      = (const float*)d_in[1];
  const float* v      = (const float*)d_in[2];
  const float* q_mask = (const float*)d_in[3];
  const float* k_mask = (const float*)d_in[4];
  const float* Wq = (const float*)d_in[5];
  const float* bq = (const float*)d_in[6];
  const float* Wk = (const float*)d_in[7];
  const float* bk = (const float*)d_in[8];
  const float* Wv = (const float*)d_in[9];
  const float* bv = (const float*)d_in[10];
  const float* Wf = (const float*)d_in[11];
  const float* bf = (const float*)d_in[12];
  const float* gamma = (const float*)d_in[13];
  const float* beta  = (const float*)d_in[14];

  float* ws = (float*)d_ws;
  const size_t PLANE = (size_t)B_ * H_ * S_ * DO_;   // 4M floats
  float* Qh     = ws;
  float* Kh     = ws + PLANE;
  float* Vh     = ws + 2 * PLANE;
  float* ctx    = ws + 3 * PLANE;
  float* rowmax = ws + 4 * PLANE;
  float* rowsum = rowmax + (size_t)B_ * H_ * S_;

  float* out  = (float*)d_out;
  float* attn = out + (size_t)B_ * S_ * D_;          // tuple: [out | attention]

  const dim3 gGemm(128, 4, 1);                       // (M/32, N/256)
  gemm_xw<<<gGemm, 128, 0, stream>>>(q, Wq, bq, Qh, 1);
  gemm_xw<<<gGemm, 128, 0, stream>>>(k, Wk, bk, Kh, 1);
  gemm_xw<<<gGemm, 128, 0, stream>>>(v, Wv, bv, Vh, 1);

  attn_scores <<<dim3(32, 4, 64), 128, 0, stream>>>(Qh, Kh, attn);
  softmax_stats<<<8192, 256, 0, stream>>>(attn, q_mask, k_mask, rowmax, rowsum);
  attn_context<<<dim3(8, 1, 64), 128, 0, stream>>>(attn, Vh, q_mask, k_mask,
                                                   rowmax, rowsum, ctx);
  gemm_xw<<<gGemm, 128, 0, stream>>>(ctx, Wf, bf, out, 0);
  residual_ln<<<512, 256, 0, stream>>>(out, v, gamma, beta);
}